// FeatureBatchSpatialDirectionalGraphConv_34772055229049
// MI455X (gfx1250) — compile-verified
//
#include <hip/hip_runtime.h>
#include <hip/hip_bf16.h>

typedef float v2f __attribute__((ext_vector_type(2)));
typedef float v8f __attribute__((ext_vector_type(8)));

#define D_IN   256
#define D_HALF 128

// ---------------------------------------------------------------------------
// 1) out[n, j] = bias[j]   (scatter accumulates on top -> bias fused)
//    degree arrays start at 1.0f (self-loop contribution)
// ---------------------------------------------------------------------------
__global__ void k_init(float* __restrict__ out, const float* __restrict__ bias,
                       float* __restrict__ deg_in, float* __restrict__ deg_out,
                       int n_nodes) {
  long long idx = (long long)blockIdx.x * blockDim.x + threadIdx.x;
  long long total = (long long)n_nodes * D_IN;
  if (idx < total) out[idx] = bias[idx & (D_IN - 1)];
  if (idx < n_nodes) { deg_in[idx] = 1.0f; deg_out[idx] = 1.0f; }
}

// ---------------------------------------------------------------------------
// 2) degree histogram: in_deg over ei[0] (col), out_deg over ei[1] (row)
// ---------------------------------------------------------------------------
__global__ void k_degree(const long long* __restrict__ ei,
                         float* __restrict__ deg_in, float* __restrict__ deg_out,
                         int n_edges) {
  int e = blockIdx.x * blockDim.x + threadIdx.x;
  if (e >= n_edges) return;
  atomicAdd(&deg_in[(int)ei[e]], 1.0f);
  atomicAdd(&deg_out[(int)ei[(long long)n_edges + e]], 1.0f);
}

// ---------------------------------------------------------------------------
// 3) deg -> rsqrt(deg)   (deg >= 1 always, matches reference's inf->0 guard)
// ---------------------------------------------------------------------------
__global__ void k_rsqrt(float* __restrict__ deg_in, float* __restrict__ deg_out, int n) {
  int i = blockIdx.x * blockDim.x + threadIdx.x;
  if (i < n) { deg_in[i] = rsqrtf(deg_in[i]); deg_out[i] = rsqrtf(deg_out[i]); }
}

// ---------------------------------------------------------------------------
// 4) Fused projection GEMM with fp32 WMMA (V_WMMA_F32_16X16X4_F32):
//    xall[n, j] = sum_k x[n,k] * W[j,k],  W = [W_f ; W_b] (rows 0..127 / 128..255)
//    Block = 512 threads = 16 waves; block handles a 16-row x 256-col slab.
//    Wave w computes the 16x16 tile at columns [16w, 16w+16).
//
//    A frag (16x4 f32, 2 VGPR): lane l -> A[l&15][k0 + 2*(l>>4) + {0,1}]
//    B frag (4x16  f32, 2 VGPR): lane l -> B[k0 + 2*(l>>4) + {0,1}][l&15]
//                                        = W[l&15 + n_base][k0 + 2*(l>>4) + {0,1}]
//    C/D  (16x16 f32, 8 VGPR): reg r -> row r + 8*(lane>>4), col lane&15
// ---------------------------------------------------------------------------
__global__ void __launch_bounds__(512) k_gemm_wmma(
    const float* __restrict__ x, const float* __restrict__ Wf,
    const float* __restrict__ Wb, float* __restrict__ xall) {
  const int wave   = threadIdx.x >> 5;
  const int lane   = threadIdx.x & 31;
  const int m_base = blockIdx.x << 4;
  const int n_base = wave << 4;
  const int t      = lane & 15;
  const int koff   = (lane >> 4) << 1;   // 0 or 2

  const float* arow = x + (long long)(m_base + t) * D_IN + koff;
  const float* wrow = (n_base < D_HALF
                           ? Wf + (long long)(n_base + t) * D_IN
                           : Wb + (long long)(n_base - D_HALF + t) * D_IN) + koff;

  v8f c = {0.f, 0.f, 0.f, 0.f, 0.f, 0.f, 0.f, 0.f};
#pragma unroll 8
  for (int k0 = 0; k0 < D_IN; k0 += 4) {
    v2f a = *(const v2f*)(arow + k0);   // 8B-aligned b64 load
    v2f b = *(const v2f*)(wrow + k0);
    c = __builtin_amdgcn_wmma_f32_16x16x4_f32(
        /*neg_a=*/false, a, /*neg_b=*/false, b,
        /*c_mod=*/(short)0, c, /*reuse_a=*/false, /*reuse_b=*/false);
  }

  const int mrow = m_base + ((lane >> 4) << 3);
  float* o = xall + (long long)mrow * D_IN + n_base + t;
#pragma unroll
  for (int r = 0; r < 8; ++r) o[(long long)r * D_IN] = c[r];
}

// Scalar tail for n_nodes not multiple of 16 (unused for N=100000, kept for safety)
__global__ void k_gemm_tail(const float* __restrict__ x, const float* __restrict__ Wf,
                            const float* __restrict__ Wb, float* __restrict__ xall,
                            int row0, int n_rows) {
  int n = row0 + blockIdx.x;
  if (blockIdx.x >= n_rows) return;
  int j = threadIdx.x;  // 0..255
  const float* w = (j < D_HALF) ? (Wf + (long long)j * D_IN)
                                : (Wb + (long long)(j - D_HALF) * D_IN);
  const float* xr = x + (long long)n * D_IN;
  float acc = 0.f;
  for (int k = 0; k < D_IN; ++k) acc = fmaf(xr[k], w[k], acc);
  xall[(long long)n * D_IN + j] = acc;
}

// ---------------------------------------------------------------------------
// 5) Edge scatter (one wave per edge, self-loops appended at e >= n_edges):
//    w = rsqrt(out_deg[row]) * rsqrt(in_deg[col])
//    out[col, 0:128]   += xall[row, 0:128]   * w   (forward)
//    out[row, 128:256] += xall[col, 128:256] * w   (backward)
// ---------------------------------------------------------------------------
__global__ void k_scatter(const long long* __restrict__ ei, const float* __restrict__ xall,
                          const float* __restrict__ inv_in, const float* __restrict__ inv_out,
                          float* __restrict__ out, int n_edges, int n_nodes) {
  int e    = (blockIdx.x << 3) + (threadIdx.x >> 5);
  int lane = threadIdx.x & 31;
  if (e >= n_edges + n_nodes) return;

  int c, r;
  if (e < n_edges) {
    c = (int)ei[e];                        // edge_index[0] = col
    r = (int)ei[(long long)n_edges + e];   // edge_index[1] = row
  } else {
    c = r = e - n_edges;                   // self-loop
  }

  const float w = inv_out[r] * inv_in[c];
  const float* xf = xall + (long long)r * D_IN;            // forward msg source
  const float* xb = xall + (long long)c * D_IN + D_HALF;   // backward msg source
  float* of = out + (long long)c * D_IN;
  float* ob = out + (long long)r * D_IN + D_HALF;

  __builtin_prefetch(xf + lane, 0, 1);   // global_prefetch_b8
  __builtin_prefetch(xb + lane, 0, 1);

#pragma unroll
  for (int k = 0; k < D_HALF; k += 32) {
    atomicAdd(&of[k + lane], xf[k + lane] * w);
    atomicAdd(&ob[k + lane], xb[k + lane] * w);
  }
}

// ---------------------------------------------------------------------------
extern "C" void kernel_launch(void* const* d_in, const int* in_sizes, int n_in,
                              void* d_out, int out_size, void* d_ws, size_t ws_size,
                              hipStream_t stream) {
  const float*     x    = (const float*)d_in[0];
  const long long* ei   = (const long long*)d_in[1];   // int64 [2, E]
  const float*     Wf   = (const float*)d_in[2];
  const float*     Wb   = (const float*)d_in[3];
  const float*     bias = (const float*)d_in[4];

  const int n_nodes = in_sizes[0] / D_IN;
  const int n_edges = in_sizes[1] / 2;

  float* out     = (float*)d_out;
  float* xall    = (float*)d_ws;                         // [N, 256]
  float* deg_in  = xall + (size_t)n_nodes * D_IN;        // [N]
  float* deg_out = deg_in + n_nodes;                     // [N]

  // 1) init out = bias, degrees = 1
  {
    long long total = (long long)n_nodes * D_IN;
    int blocks = (int)((total + 255) / 256);
    k_init<<<blocks, 256, 0, stream>>>(out, bias, deg_in, deg_out, n_nodes);
  }
  // 2) degree histogram
  k_degree<<<(n_edges + 255) / 256, 256, 0, stream>>>(ei, deg_in, deg_out, n_edges);
  // 3) rsqrt
  k_rsqrt<<<(n_nodes + 255) / 256, 256, 0, stream>>>(deg_in, deg_out, n_nodes);
  // 4) fused projection GEMM via WMMA
  {
    int mtiles = n_nodes >> 4;
    if (mtiles > 0)
      k_gemm_wmma<<<mtiles, 512, 0, stream>>>(x, Wf, Wb, xall);
    int tail = n_nodes - (mtiles << 4);
    if (tail > 0)
      k_gemm_tail<<<tail, 256, 0, stream>>>(x, Wf, Wb, xall, mtiles << 4, tail);
  }
  // 5) edge + self-loop scatter
  {
    int total_e = n_edges + n_nodes;
    k_scatter<<<(total_e + 7) / 8, 256, 0, stream>>>(ei, xall, deg_in, deg_out,
                                                     out, n_edges, n_nodes);
  }
}